// SwinTransformer_8933531976313
// MI455X (gfx1250) — compile-verified
//
#include <hip/hip_runtime.h>
#include <math.h>

// ---------------- types for CDNA5 f32 WMMA (16x16x4) ----------------
typedef float v2f __attribute__((ext_vector_type(2)));
typedef float v4f __attribute__((ext_vector_type(4)));
typedef float v8f __attribute__((ext_vector_type(8)));

// Model dims
#define C_ 384
#define NH_ 12
#define HD_ 32
#define L_ 256
#define MLP_ 1536
#define QKV_ 1152

// ---------------------------------------------------------------
// Generic f32 WMMA GEMM: Out[n, m] = act( sum_k W[m,k]*X[n,k] + C[n,m] + bias[m] )
// W: (M x K) row-major, X: (N x K) row-major, Out/Cin: (N x M) row-major.
// One wave -> one 16x16 tile.  A/B lane layout per CDNA5 ISA 16x4 striping:
// lane L holds rows (L&15); VGPR v holds K = kbase + 2*(L>>4) + v.
// ---------------------------------------------------------------
template<bool GELU, bool ADDC>
__global__ void gemm_atb(const float* __restrict__ W, const float* __restrict__ X,
                         const float* __restrict__ bias, const float* __restrict__ Cin,
                         float* __restrict__ Out, int M, int N, int K) {
    int wv = threadIdx.x >> 5;
    int tile = blockIdx.x * 4 + wv;
    int tilesM = M >> 4;
    if (tile >= tilesM * (N >> 4)) return;   // wave-uniform, EXEC stays full
    int tm = tile % tilesM;
    int tn = tile / tilesM;
    int lane = threadIdx.x & 31;
    int r16  = lane & 15;
    int khi  = (lane >> 4) << 1;

    const float* Ap = W + (size_t)(tm * 16 + r16) * K + khi;
    const float* Bp = X + (size_t)(tn * 16 + r16) * K + khi;

    int nrow  = tn * 16 + r16;                 // N index of this lane in C/D
    int mbase = tm * 16 + ((lane >> 4) << 3);  // M base (8 consecutive)

    v8f acc = {0.f, 0.f, 0.f, 0.f, 0.f, 0.f, 0.f, 0.f};
    if (ADDC) {
        const float* cp = Cin + (size_t)nrow * M + mbase;
        v4f c0 = *(const v4f*)cp;
        v4f c1 = *(const v4f*)(cp + 4);
        acc[0] = c0.x; acc[1] = c0.y; acc[2] = c0.z; acc[3] = c0.w;
        acc[4] = c1.x; acc[5] = c1.y; acc[6] = c1.z; acc[7] = c1.w;
    }
#pragma unroll 4
    for (int k = 0; k < K; k += 4) {
        v2f a = *(const v2f*)(Ap + k);
        v2f b = *(const v2f*)(Bp + k);
        acc = __builtin_amdgcn_wmma_f32_16x16x4_f32(false, a, false, b,
                                                    (short)0, acc, false, false);
    }
    float* op = Out + (size_t)nrow * M + mbase;
#pragma unroll
    for (int j = 0; j < 8; j++) {
        float v = acc[j] + (bias ? bias[mbase + j] : 0.0f);
        if (GELU) v = 0.5f * v * (1.0f + erff(v * 0.70710678118654752f));
        op[j] = v;
    }
}

// ---------------------------------------------------------------
// Patch embed (conv k=16 s=16) fused with buggy-flatten + pos_embed.
// The reference's "buggy" flatten is a raw reinterpret of the contiguous
// (C,16,16) feature buffer as (L,C), so we write flat index i directly.
// ---------------------------------------------------------------
__global__ void patch_embed_kernel(const float* __restrict__ x, const float* __restrict__ pw,
                                   const float* __restrict__ pb, const float* __restrict__ pos,
                                   float* __restrict__ tok) {
    int i = blockIdx.x * 256 + threadIdx.x;     // 0..98303 : flat (C,16,16) index
    int c  = i >> 8;
    int sp = i & 255;
    int hp = sp >> 4, wp = sp & 15;
    const float* wrow = pw + c * 768;
    float s = pb[c];
    for (int ic = 0; ic < 3; ic++)
        for (int p = 0; p < 16; p++) {
            const float* xr = x + ic * 65536 + (hp * 16 + p) * 256 + wp * 16;
            const float* wr = wrow + ic * 256 + p * 16;
#pragma unroll 4
            for (int q = 0; q < 16; q++) s += xr[q] * wr[q];
        }
    tok[i] = s + pos[i];
}

// ---------------------------------------------------------------
// LayerNorm, one wave32 per token row.
// mode 0: out[row] = LN(h[row])       (plain, for ln2)
// mode 1: out[window-ordered row] = LN(h[rolled source row])  (ln1 + shift + partition)
// ---------------------------------------------------------------
__global__ void ln_kernel(const float* __restrict__ h, const float* __restrict__ w,
                          const float* __restrict__ b, float* __restrict__ out, int mode) {
    int row  = blockIdx.x;       // 0..255 output row
    int lane = threadIdx.x;      // 32 lanes
    int src  = row;
    if (mode == 1) {
        int wd = row >> 6, i = row & 63;
        int wr = wd >> 1, wc = wd & 1, ir = i >> 3, ic = i & 7;
        int rp = wr * 8 + ir, cp = wc * 8 + ic;
        src = (((rp + 4) & 15) << 4) + ((cp + 4) & 15);   // roll(-4,-4)
    }
    const float* x = h + src * C_;
    float vals[12];
    float s = 0.f;
#pragma unroll
    for (int t = 0; t < 12; t++) { vals[t] = x[lane + 32 * t]; s += vals[t]; }
    for (int off = 16; off; off >>= 1) s += __shfl_xor(s, off, 32);
    float mean = s * (1.0f / C_);
    float vs = 0.f;
#pragma unroll
    for (int t = 0; t < 12; t++) { float d = vals[t] - mean; vs += d * d; }
    for (int off = 16; off; off >>= 1) vs += __shfl_xor(vs, off, 32);
    float inv = rsqrtf(vs * (1.0f / C_) + 1e-5f);
    float* o = out + row * C_;
#pragma unroll
    for (int t = 0; t < 12; t++) {
        int c = lane + 32 * t;
        o[c] = (vals[t] - mean) * inv * w[c] + b[c];
    }
}

// ---------------------------------------------------------------
// Windowed attention for one (window, head): S = qk^T*scale + relbias + mask,
// softmax rows (LDS), O = S @ V.  All matmuls via f32 WMMA.
// qkv layout: [token n (window-ordered)][part*384 + head*32 + d], n-major.
// ---------------------------------------------------------------
__global__ void attn_kernel(const float* __restrict__ qkv, const float* __restrict__ rpb,
                            float* __restrict__ obuf) {
    __shared__ float S[64 * 66];
    int wd = blockIdx.x / NH_;
    int hd = blockIdx.x % NH_;
    int wv   = threadIdx.x >> 5;
    int lane = threadIdx.x & 31;
    int r16  = lane & 15;
    int khi  = (lane >> 4) << 1;
    const float scale = 0.17677669529663687f;   // 1/sqrt(32)

    const float* qb = qkv + (size_t)(wd * 64) * QKV_ + hd * HD_;
    const float* kb = qb + C_;
    const float* vb = qb + 2 * C_;
    int wr = wd >> 1, wc = wd & 1;

    // ---- S = q k^T  (M=64 i, N=64 j, K=32 d): 16 tiles over 4 waves ----
    for (int t = 0; t < 4; t++) {
        int tile = wv * 4 + t;
        int tm = tile >> 2, tn = tile & 3;
        v8f acc = {0.f, 0.f, 0.f, 0.f, 0.f, 0.f, 0.f, 0.f};
        const float* Ap = qb + (size_t)(tm * 16 + r16) * QKV_ + khi;
        const float* Bp = kb + (size_t)(tn * 16 + r16) * QKV_ + khi;
#pragma unroll
        for (int k = 0; k < HD_; k += 4) {
            v2f a = *(const v2f*)(Ap + k);
            v2f b = *(const v2f*)(Bp + k);
            acc = __builtin_amdgcn_wmma_f32_16x16x4_f32(false, a, false, b,
                                                        (short)0, acc, false, false);
        }
        int j = tn * 16 + r16;
        int jr = j >> 3, jc = j & 7;
        int gjr = wr * 8 + jr, gjc = wc * 8 + jc;
        int mj = ((gjr < 8) ? 0 : ((gjr < 12) ? 1 : 2)) * 3 +
                 ((gjc < 8) ? 0 : ((gjc < 12) ? 1 : 2));
        int ibase = tm * 16 + ((lane >> 4) << 3);
#pragma unroll
        for (int u = 0; u < 8; u++) {
            int i = ibase + u;
            int ir = i >> 3, ic = i & 7;
            int gir = wr * 8 + ir, gic = wc * 8 + ic;
            int mi = ((gir < 8) ? 0 : ((gir < 12) ? 1 : 2)) * 3 +
                     ((gic < 8) ? 0 : ((gic < 12) ? 1 : 2));
            int rel = (ir - jr + 7) * 15 + (ic - jc + 7);
            float bias = rpb[rel * NH_ + hd];
            float mask = (mi != mj) ? -100.0f : 0.0f;
            S[i * 66 + j] = acc[u] * scale + bias + mask;
        }
    }
    __syncthreads();

    // ---- row softmax in LDS ----
    if (threadIdx.x < 64) {
        float* row = S + threadIdx.x * 66;
        float mx = row[0];
        for (int j = 1; j < 64; j++) mx = fmaxf(mx, row[j]);
        float sum = 0.f;
        for (int j = 0; j < 64; j++) { float e = __expf(row[j] - mx); row[j] = e; sum += e; }
        float inv = 1.0f / sum;
        for (int j = 0; j < 64; j++) row[j] *= inv;
    }
    __syncthreads();

    // ---- O = P @ V  (M=64 i, N=32 d, K=64 j): 8 tiles over 4 waves ----
    for (int t = 0; t < 2; t++) {
        int tile = wv * 2 + t;
        int tm = tile >> 1, tn = tile & 1;
        v8f acc = {0.f, 0.f, 0.f, 0.f, 0.f, 0.f, 0.f, 0.f};
        const float* Ap = S + (tm * 16 + r16) * 66 + khi;
        const float* Vp = vb + tn * 16 + r16;
#pragma unroll 4
        for (int k = 0; k < 64; k += 4) {
            v2f a = *(const v2f*)(Ap + k);
            v2f b;
            b.x = Vp[(size_t)(k + khi) * QKV_];
            b.y = Vp[(size_t)(k + khi + 1) * QKV_];
            acc = __builtin_amdgcn_wmma_f32_16x16x4_f32(false, a, false, b,
                                                        (short)0, acc, false, false);
        }
        int d = tn * 16 + r16;
        int ibase = tm * 16 + ((lane >> 4) << 3);
#pragma unroll
        for (int u = 0; u < 8; u++)
            obuf[(size_t)(wd * 64 + ibase + u) * C_ + hd * HD_ + d] = acc[u];
    }
}

// h[token] += proj_out[window-ordered token after reverse-partition + roll(+4,+4)]
__global__ void add_unshift_kernel(float* __restrict__ h, const float* __restrict__ p) {
    int idx = blockIdx.x * 256 + threadIdx.x;   // 0..98303
    int l = idx / C_, c = idx % C_;
    int r = l >> 4, cc = l & 15;
    int rp = (r + 12) & 15, cp = (cc + 12) & 15;    // inverse of roll(+4)
    int wd = ((rp >> 3) << 1) + (cp >> 3);
    int i  = ((rp & 7) << 3) + (cp & 7);
    h[idx] += p[(size_t)(wd * 64 + i) * C_ + c];
}

// f_chw[c,16,16] = tokens (L,C) transposed
__global__ void transpose_cl_kernel(const float* __restrict__ tok, float* __restrict__ f) {
    int i = blockIdx.x * 256 + threadIdx.x;   // c*256 + sp
    int c = i >> 8, sp = i & 255;
    f[i] = tok[sp * C_ + c];
}

// skip = 1x1 conv C->4
__global__ void skip_kernel(const float* __restrict__ f, const float* __restrict__ w,
                            const float* __restrict__ b, float* __restrict__ skip) {
    int i = blockIdx.x * 256 + threadIdx.x;   // 0..1023
    int o = i >> 8, sp = i & 255;
    float s = b[o];
    const float* wr = w + o * C_;
    for (int c = 0; c < C_; c++) s += f[c * 256 + sp] * wr[c];
    skip[i] = s;
}

// dt1 (transpose-conv as einsum) + BN(identity stats) + ReLU -> (3,256,256)
__global__ void dt1_kernel(const float* __restrict__ f, const float* __restrict__ w,
                           const float* __restrict__ b, const float* __restrict__ bnw,
                           const float* __restrict__ bnb, float* __restrict__ d1) {
    int i = blockIdx.x * 256 + threadIdx.x;   // 0..196607
    int o = i / 65536;
    int rem = i & 65535;
    int H = rem >> 8, Wc = rem & 255;
    int hp = H >> 4, k = H & 15, wp = Wc >> 4, l = Wc & 15;
    float s = b[o];
    const float* wp_ = w + o * 256 + k * 16 + l;    // stride 768 over c
    const float* fp  = f + hp * 16 + wp;            // stride 256 over c
    for (int c = 0; c < C_; c++) s += fp[c * 256] * wp_[c * 768];
    s = s * 0.99999500003750f * bnw[o] + bnb[o];
    d1[i] = fmaxf(s, 0.0f);
}

// dt2 + BN + bilinear(align_corners) skip upsample + add -> 67,108,864 outputs
__global__ void final_kernel(const float* __restrict__ d1, const float* __restrict__ w2,
                             const float* __restrict__ b2, const float* __restrict__ bnw,
                             const float* __restrict__ bnb, const float* __restrict__ skip,
                             float* __restrict__ out) {
    unsigned int i = blockIdx.x * 256u + threadIdx.x;   // < 4*4096*4096
    int oc  = i >> 24;
    int rem = i & 0xFFFFFF;
    int H2 = rem >> 12, W2 = rem & 4095;
    int h = H2 >> 4, k = H2 & 15, w = W2 >> 4, l = W2 & 15;
    float s = b2[oc];
#pragma unroll
    for (int c = 0; c < 3; c++)
        s += d1[c * 65536 + h * 256 + w] * w2[(c * 4 + oc) * 256 + k * 16 + l];
    s = s * 0.99999500003750f * bnw[oc] + bnb[oc];
    // bilinear align_corners upsample of (16,16) skip to (4096,4096)
    float fy = H2 * (15.0f / 4095.0f);
    int y0 = (int)floorf(fy); int y1 = (y0 + 1 > 15) ? 15 : y0 + 1; float wy = fy - y0;
    float fx = W2 * (15.0f / 4095.0f);
    int x0 = (int)floorf(fx); int x1 = (x0 + 1 > 15) ? 15 : x0 + 1; float wx = fx - x0;
    const float* sk = skip + oc * 256;
    float r0 = sk[y0 * 16 + x0] * (1.0f - wy) + sk[y1 * 16 + x0] * wy;
    float r1 = sk[y0 * 16 + x1] * (1.0f - wy) + sk[y1 * 16 + x1] * wy;
    out[i] = s + r0 * (1.0f - wx) + r1 * wx;
}

// ---------------------------------------------------------------
extern "C" void kernel_launch(void* const* d_in, const int* in_sizes, int n_in,
                              void* d_out, int out_size, void* d_ws, size_t ws_size,
                              hipStream_t stream) {
    (void)in_sizes; (void)n_in; (void)out_size; (void)ws_size;
    const float* x       = (const float*)d_in[0];
    const float* patch_w = (const float*)d_in[1];
    const float* patch_b = (const float*)d_in[2];
    const float* pos     = (const float*)d_in[3];
    const float* ln1w    = (const float*)d_in[4];
    const float* ln1b    = (const float*)d_in[5];
    const float* qkvw    = (const float*)d_in[6];
    const float* projw   = (const float*)d_in[7];
    const float* projb   = (const float*)d_in[8];
    const float* rpb     = (const float*)d_in[9];
    const float* ln2w    = (const float*)d_in[10];
    const float* ln2b    = (const float*)d_in[11];
    const float* fc1w    = (const float*)d_in[12];
    const float* fc1b    = (const float*)d_in[13];
    const float* fc2w    = (const float*)d_in[14];
    const float* fc2b    = (const float*)d_in[15];
    const float* skipw   = (const float*)d_in[16];
    const float* skipb   = (const float*)d_in[17];
    const float* dt1w    = (const float*)d_in[18];
    const float* dt1b    = (const float*)d_in[19];
    const float* bn1w    = (const float*)d_in[20];
    const float* bn1b    = (const float*)d_in[21];
    const float* dt2w    = (const float*)d_in[22];
    const float* dt2b    = (const float*)d_in[23];
    const float* bn2w    = (const float*)d_in[24];
    const float* bn2b    = (const float*)d_in[25];

    float* ws    = (float*)d_ws;
    float* tok   = ws;                 // 98304   tokens (L,C) == flat (C,16,16)
    float* ybuf  = tok   + 98304;      // 98304   LN1 + shifted-window rows
    float* qkvb  = ybuf  + 98304;      // 294912  (256 x 1152) token-major
    float* obuf  = qkvb  + 294912;     // 98304   attention output
    float* pbuf  = obuf  + 98304;      // 98304   proj output (window-ordered)
    float* zbuf  = pbuf  + 98304;      // 98304   LN2 output
    float* h1    = zbuf  + 98304;      // 393216  fc1/gelu output (256 x 1536)
    float* fchw  = h1    + 393216;     // 98304   (C,16,16) decoder input
    float* skb   = fchw  + 98304;      // 1024    skip (4,16,16)
    float* d1    = skb   + 1024;       // 196608  (3,256,256)

    patch_embed_kernel<<<384, 256, 0, stream>>>(x, patch_w, patch_b, pos, tok);

    for (int blk = 0; blk < 12; blk++) {
        ln_kernel<<<256, 32, 0, stream>>>(tok, ln1w + blk * 384, ln1b + blk * 384, ybuf, 1);
        // qkv: (256 x 1152) = ybuf(256x384) . qkv_w^T ; 1152 tiles / 4 waves
        gemm_atb<false, false><<<288, 128, 0, stream>>>(
            qkvw + (size_t)blk * 442368, ybuf, nullptr, nullptr, qkvb, QKV_, L_, C_);
        attn_kernel<<<48, 128, 0, stream>>>(qkvb, rpb + blk * 2700, obuf);
        // proj: 384 tiles / 4 waves
        gemm_atb<false, false><<<96, 128, 0, stream>>>(
            projw + (size_t)blk * 147456, obuf, projb + blk * 384, nullptr, pbuf, C_, L_, C_);
        add_unshift_kernel<<<384, 256, 0, stream>>>(tok, pbuf);
        ln_kernel<<<256, 32, 0, stream>>>(tok, ln2w + blk * 384, ln2b + blk * 384, zbuf, 0);
        // fc1 + GELU: 1536 tiles / 4 waves
        gemm_atb<true, false><<<384, 128, 0, stream>>>(
            fc1w + (size_t)blk * 589824, zbuf, fc1b + blk * 1536, nullptr, h1, MLP_, L_, C_);
        // fc2 + residual (h preloaded as WMMA C operand): 384 tiles / 4 waves
        gemm_atb<false, true><<<96, 128, 0, stream>>>(
            fc2w + (size_t)blk * 589824, h1, fc2b + blk * 384, tok, tok, C_, L_, MLP_);
    }

    transpose_cl_kernel<<<384, 256, 0, stream>>>(tok, fchw);
    skip_kernel<<<4, 256, 0, stream>>>(fchw, skipw, skipb, skb);
    dt1_kernel<<<768, 256, 0, stream>>>(fchw, dt1w, dt1b, bn1w, bn1b, d1);
    final_kernel<<<262144, 256, 0, stream>>>(d1, dt2w, dt2b, bn2w, bn2b, skb, (float*)d_out);
}